// VectorQuantizer_17179869990
// MI455X (gfx1250) — compile-verified
//
#include <hip/hip_runtime.h>
#include <hip/hip_bf16.h>
#include <math.h>

// Problem constants
#define NROWS 16384      // 8*2048 tokens
#define DIM   512        // embedding dim
#define KCODE 8192       // codebook entries
// Tiling
#define BM 64            // rows per block
#define BN 64            // codes per chunk
#define BK 32            // d-slice per LDS stage (double-buffered)
#define LDSS (BK + 4)    // 36 floats: 144B rows -> 16B aligned, conflict-free b64 reads

typedef float v2f __attribute__((ext_vector_type(2)));
typedef float v8f __attribute__((ext_vector_type(8)));

// ---------------------------------------------------------------------------
// CDNA5 async Global->LDS copy (ASYNCcnt path), with sync fallback.
#if defined(__has_builtin)
#if __has_builtin(__builtin_amdgcn_global_load_async_to_lds_b128)
#define VQ_HAVE_ASYNC_LDS 1
#endif
#endif

// Parameter types per hipcc diagnostic: int32x4 pointers, AS1 src / AS3 dst.
typedef int vq_v4i __attribute__((vector_size(16)));
typedef __attribute__((address_space(1))) vq_v4i* vq_gv4p;
typedef __attribute__((address_space(3))) vq_v4i* vq_lv4p;

__device__ __forceinline__ void vq_copy_b128(const float* __restrict__ g,
                                             float* __restrict__ l) {
#if defined(VQ_HAVE_ASYNC_LDS)
    __builtin_amdgcn_global_load_async_to_lds_b128((vq_gv4p)(g), (vq_lv4p)(l), 0, 0);
#else
    *reinterpret_cast<float4*>(l) = *reinterpret_cast<const float4*>(g);
#endif
}

__device__ __forceinline__ void vq_wait_async() {
#if defined(VQ_HAVE_ASYNC_LDS)
#if __has_builtin(__builtin_amdgcn_s_wait_asynccnt)
    __builtin_amdgcn_s_wait_asynccnt(0);
#else
    asm volatile("s_wait_asynccnt 0x0" ::: "memory");
#endif
#endif
}

// ---------------------------------------------------------------------------
// Zero fill (float2, 8B-aligned region)
__global__ __launch_bounds__(256) void vq_zero_f2(float* __restrict__ p, size_t n2) {
    float2 z; z.x = 0.f; z.y = 0.f;
    size_t i = (size_t)blockIdx.x * blockDim.x + threadIdx.x;
    size_t stride = (size_t)gridDim.x * blockDim.x;
    float2* p2 = reinterpret_cast<float2*>(p);
    for (; i < n2; i += stride) p2[i] = z;
}

__global__ __launch_bounds__(256) void vq_zero_i(int* __restrict__ p, int n) {
    int i = blockIdx.x * blockDim.x + threadIdx.x;
    if (i < n) p[i] = 0;
}

// ---------------------------------------------------------------------------
// normE[k] = sum_d codebook[k][d]^2 ; one wave (32 lanes) per code
__global__ __launch_bounds__(256) void vq_norm_kernel(const float* __restrict__ cb,
                                                      float* __restrict__ normE) {
    const int lane = threadIdx.x & 31;
    const int wave = threadIdx.x >> 5;
    const int code = blockIdx.x * 8 + wave;
    const float* row = cb + (size_t)code * DIM;
    float s = 0.f;
    #pragma unroll
    for (int it = 0; it < DIM / 128; ++it) {
        float4 v = reinterpret_cast<const float4*>(row)[lane + it * 32];
        s += v.x * v.x + v.y * v.y + v.z * v.z + v.w * v.w;
    }
    #pragma unroll
    for (int off = 16; off > 0; off >>= 1) s += __shfl_xor(s, off, 32);
    if (lane == 0) normE[code] = s;
}

// ---------------------------------------------------------------------------
// Main kernel: distances via V_WMMA_F32_16X16X4_F32, running argmin per row.
// 256 threads = 8 waves in 4(row)x2(code) arrangement; wave tile 16x32.
// Double-buffered LDS stages filled by async Global->LDS DMA.
__global__ __launch_bounds__(256) void vq_argmin_kernel(
    const float* __restrict__ x,      // [NROWS][DIM]
    const float* __restrict__ cb,     // [KCODE][DIM]
    const float* __restrict__ normE,  // [KCODE]
    int* __restrict__ idxOut)         // [NROWS]
{
    __shared__ float As[2][BM][LDSS];
    __shared__ float Bs[2][BN][LDSS];
    __shared__ float mergeV[2][BM];
    __shared__ int   mergeI[2][BM];

    const int tid  = threadIdx.x;
    const int lane = tid & 31;
    const int wave = tid >> 5;        // 0..7
    const int wr   = wave >> 1;       // 0..3 : row group (16 rows each)
    const int wc   = wave & 1;        // 0..1 : code half (32 codes each)
    const int h    = lane >> 4;       // lane half
    const int ln   = lane & 15;
    const int rowBase = blockIdx.x * BM;
    const int m0 = wr * 16;
    const int n0 = wc * 32;

    // Stage-loader indices: 256 threads x 2 b128 per matrix tile (64x32 floats)
    const int lr = tid >> 2;          // 0..63 row within tile
    const int lq = (tid & 3) * 8;     // starting float column (quarter of 32)

    float minv[8];
    int   mini[8];
    #pragma unroll
    for (int i = 0; i < 8; ++i) { minv[i] = 3.4e38f; mini[i] = 0; }

    // Prologue: stage (chunk=0, d0=0) into buffer 0
    #pragma unroll
    for (int j = 0; j < 2; ++j) {
        const int col = lq + j * 4;
        vq_copy_b128(&x [(size_t)(rowBase + lr) * DIM + col], &As[0][lr][col]);
        vq_copy_b128(&cb[(size_t)lr            * DIM + col], &Bs[0][lr][col]);
    }

    int buf = 0;
    for (int chunk = 0; chunk < KCODE; chunk += BN) {
        v8f acc0 = {0.f,0.f,0.f,0.f,0.f,0.f,0.f,0.f};
        v8f acc1 = {0.f,0.f,0.f,0.f,0.f,0.f,0.f,0.f};

        for (int d0 = 0; d0 < DIM; d0 += BK) {
            vq_wait_async();
            __syncthreads();

            // Issue async stage of the next tile into the other buffer
            int nchunk = chunk, nd0 = d0 + BK;
            if (nd0 == DIM) { nd0 = 0; nchunk += BN; }
            if (nchunk < KCODE) {
                const int nb = buf ^ 1;
                #pragma unroll
                for (int j = 0; j < 2; ++j) {
                    const int col = lq + j * 4;
                    vq_copy_b128(&x [(size_t)(rowBase + lr) * DIM + nd0 + col],
                                 &As[nb][lr][col]);
                    vq_copy_b128(&cb[(size_t)(nchunk  + lr) * DIM + nd0 + col],
                                 &Bs[nb][lr][col]);
                }
            }

            // Compute on current buffer
            #pragma unroll
            for (int kk = 0; kk < BK; kk += 4) {
                const int kc = kk + 2 * h;
                v2f a, b0, b1;
                a.x  = As[buf][m0 + ln][kc];       a.y  = As[buf][m0 + ln][kc + 1];
                b0.x = Bs[buf][n0 + ln][kc];       b0.y = Bs[buf][n0 + ln][kc + 1];
                b1.x = Bs[buf][n0 + 16 + ln][kc];  b1.y = Bs[buf][n0 + 16 + ln][kc + 1];
                acc0 = __builtin_amdgcn_wmma_f32_16x16x4_f32(
                    false, a, false, b0, (short)0, acc0, false, false);
                acc1 = __builtin_amdgcn_wmma_f32_16x16x4_f32(
                    false, a, false, b1, (short)0, acc1, false, false);
            }
            buf ^= 1;
        }

        // distances for this chunk: ||e||^2 - 2*dot  (||x||^2 constant per row)
        const int code0 = chunk + n0 + ln;
        const int code1 = code0 + 16;
        const float ne0 = normE[code0];
        const float ne1 = normE[code1];

        #pragma unroll
        for (int i = 0; i < 8; ++i) {
            {
                float v = fmaf(-2.0f, acc0[i], ne0);
                int   c = code0;
                #pragma unroll
                for (int off = 1; off < 16; off <<= 1) {
                    float ov = __shfl_xor(v, off, 16);
                    int   oc = __shfl_xor(c, off, 16);
                    if (ov < v || (ov == v && oc < c)) { v = ov; c = oc; }
                }
                if (v < minv[i]) { minv[i] = v; mini[i] = c; }
            }
            {
                float v = fmaf(-2.0f, acc1[i], ne1);
                int   c = code1;
                #pragma unroll
                for (int off = 1; off < 16; off <<= 1) {
                    float ov = __shfl_xor(v, off, 16);
                    int   oc = __shfl_xor(c, off, 16);
                    if (ov < v || (ov == v && oc < c)) { v = ov; c = oc; }
                }
                if (v < minv[i]) { minv[i] = v; mini[i] = c; }
            }
        }
    }

    // merge the two code-half waves per row group
    __syncthreads();
    if (ln == 0) {   // lanes 0 and 16
        #pragma unroll
        for (int i = 0; i < 8; ++i) {
            const int m = m0 + h * 8 + i;     // C-layout: M = i + 8h
            mergeV[wc][m] = minv[i];
            mergeI[wc][m] = mini[i];
        }
    }
    __syncthreads();
    if (tid < BM) {
        float v0 = mergeV[0][tid]; int i0 = mergeI[0][tid];
        float v1 = mergeV[1][tid]; int i1 = mergeI[1][tid];
        bool take1 = (v1 < v0) || (v1 == v0 && i1 < i0);
        idxOut[rowBase + tid] = take1 ? i1 : i0;
    }
}

// ---------------------------------------------------------------------------
// Gather quantized rows, straight-through output, SSE partials, one-hot, counts
__global__ __launch_bounds__(256) void vq_gather_kernel(
    const float* __restrict__ x, const float* __restrict__ cb,
    const int* __restrict__ idx,
    float* __restrict__ qst,     // d_out + 1
    float* __restrict__ enc,     // one-hot region (pre-zeroed)
    int* __restrict__ counts,    // [KCODE] (pre-zeroed)
    float* __restrict__ partials)// [gridDim.x]
{
    __shared__ float red[256];
    const int tid = threadIdx.x;
    const int rowBase = blockIdx.x * 64;
    float sse = 0.f;

    for (int r = 0; r < 64; ++r) {
        const int row = rowBase + r;
        const int k = idx[row];
        #pragma unroll
        for (int j = 0; j < 2; ++j) {
            const int d = tid + j * 256;
            const float q  = cb[(size_t)k * DIM + d];
            const float xi = x[(size_t)row * DIM + d];
            const float diff = q - xi;                 // matches stop_gradient(q - x)
            qst[(size_t)row * DIM + d] = xi + diff;    // straight-through forward value
            sse += diff * diff;
        }
    }
    if (tid < 64) {
        const int row = rowBase + tid;
        const int k = idx[row];
        enc[(size_t)row * KCODE + k] = 1.0f;
        atomicAdd(&counts[k], 1);
    }
    red[tid] = sse;
    __syncthreads();
    #pragma unroll
    for (int s = 128; s > 0; s >>= 1) {
        if (tid < s) red[tid] += red[tid + s];
        __syncthreads();
    }
    if (tid == 0) partials[blockIdx.x] = red[0];
}

// ---------------------------------------------------------------------------
// Scalars: loss = 1.25 * SSE/(N*D); perplexity = exp(-sum p log(p+1e-10))
__global__ __launch_bounds__(256) void vq_finalize_kernel(
    const float* __restrict__ partials, const int* __restrict__ counts,
    float* __restrict__ lossOut, float* __restrict__ perpOut)
{
    __shared__ float red[256];
    const int tid = threadIdx.x;
    float hsum = 0.f;
    for (int k = tid; k < KCODE; k += 256) {
        const float p = (float)counts[k] * (1.0f / (float)NROWS);
        hsum += p * logf(p + 1e-10f);
    }
    red[tid] = hsum;
    __syncthreads();
    #pragma unroll
    for (int s = 128; s > 0; s >>= 1) {
        if (tid < s) red[tid] += red[tid + s];
        __syncthreads();
    }
    if (tid == 0) {
        float sse = 0.f;
        for (int i = 0; i < 256; ++i) sse += partials[i];   // fixed order
        lossOut[0] = 1.25f * sse / ((float)NROWS * (float)DIM);
        perpOut[0] = expf(-red[0]);
    }
}

// ---------------------------------------------------------------------------
extern "C" void kernel_launch(void* const* d_in, const int* in_sizes, int n_in,
                              void* d_out, int out_size, void* d_ws, size_t ws_size,
                              hipStream_t stream) {
    const float* x  = (const float*)d_in[0];   // [8,2048,512]
    const float* cb = (const float*)d_in[1];   // [8192,512]

    float* out  = (float*)d_out;
    float* loss = out;                                   // [1]
    float* qst  = out + 1;                               // [8388608]
    float* perp = out + 1 + (size_t)NROWS * DIM;         // [1]
    float* enc  = out + 2 + (size_t)NROWS * DIM;         // [134217728], 8B-aligned

    float* normE    = (float*)d_ws;                                   // 8192 f
    int*   idx      = (int*)  ((char*)d_ws + (size_t)KCODE * 4);      // 16384 i
    int*   counts   = (int*)  ((char*)d_ws + (size_t)(KCODE + NROWS) * 4);        // 8192 i
    float* partials = (float*)((char*)d_ws + (size_t)(2 * KCODE + NROWS) * 4);    // 256 f

    // 1) zero one-hot region (float2 stores; region is 8-byte aligned)
    const size_t n2 = ((size_t)NROWS * KCODE) / 2;
    vq_zero_f2<<<4096, 256, 0, stream>>>(enc, n2);
    // 2) zero counts + partials (contiguous)
    vq_zero_i<<<(KCODE + 256 + 255) / 256, 256, 0, stream>>>(counts, KCODE + 256);
    // 3) codebook squared norms
    vq_norm_kernel<<<KCODE / 8, 256, 0, stream>>>(cb, normE);
    // 4) WMMA distance + argmin
    vq_argmin_kernel<<<NROWS / BM, 256, 0, stream>>>(x, cb, normE, idx);
    // 5) gather / straight-through / one-hot / histogram / SSE partials
    vq_gather_kernel<<<NROWS / 64, 256, 0, stream>>>(x, cb, idx, qst, enc, counts, partials);
    // 6) scalars
    vq_finalize_kernel<<<1, 256, 0, stream>>>(partials, counts, loss, perp);
}